// GATNet_54056458387549
// MI455X (gfx1250) — compile-verified
//
#include <hip/hip_runtime.h>

typedef float v2f __attribute__((ext_vector_type(2)));
typedef float v8f __attribute__((ext_vector_type(8)));

#define NEG_SLOPE 0.2f
// Fixed problem geometry (from the reference setup)
#define GAT_H   4
#define GAT_C   64
#define GAT_HC  256

// ---------- ordered-uint encoding for float atomicMax ----------
__device__ __forceinline__ unsigned f2ord(float f) {
    unsigned u = __float_as_uint(f);
    return (u & 0x80000000u) ? ~u : (u | 0x80000000u);
}
__device__ __forceinline__ float ord2f(unsigned k) {
    unsigned u = (k & 0x80000000u) ? (k & 0x7FFFFFFFu) : ~k;
    return __uint_as_float(u);
}

// ---------------------------------------------------------------
// C = act(A @ B + bias). A:[M,K] row-major, B:[K,Nc] row-major.
// One wave32 computes a 16x64 strip of C (4 accumulators) via
// V_WMMA_F32_16X16X4_F32, reusing each A fragment for 4 WMMAs.
// Out-of-range rows/cols are CLAMPED for loads (their products only
// reach output elements that are never stored), so the K-loop has no
// divergence and no per-iteration address recomputation.
// act: 0 = none, 1 = relu
// ---------------------------------------------------------------
__global__ __launch_bounds__(256) void wmma_gemm_f32(
    const float* __restrict__ A, const float* __restrict__ B,
    const float* __restrict__ bias, float* __restrict__ C,
    int M, int K, int Nc, int act)
{
    const int wave = threadIdx.x >> 5;
    const int lane = threadIdx.x & 31;
    const int l15  = lane & 15;
    const int kb   = (lane >> 4) << 1;       // K sub-offset {0,2}
    const int stripsN = (Nc + 63) >> 6;
    const int stripsM = (M + 15) >> 4;
    const int strip = blockIdx.x * 8 + wave;
    if (strip >= stripsM * stripsN) return;  // wave-uniform: EXEC stays full
    const int tm = strip / stripsN;
    const int sn = strip % stripsN;

    int mrow = tm * 16 + l15;
    if (mrow >= M) mrow = M - 1;             // clamp (outputs for this row not stored)
    const int c0 = sn * 64 + l15;
    int cc0 = c0;          if (cc0 >= Nc) cc0 = Nc - 1;
    int cc1 = c0 + 16;     if (cc1 >= Nc) cc1 = Nc - 1;
    int cc2 = c0 + 32;     if (cc2 >= Nc) cc2 = Nc - 1;
    int cc3 = c0 + 48;     if (cc3 >= Nc) cc3 = Nc - 1;

    const float* ap = A + (size_t)mrow * K + kb;
    const float* bp = B + (size_t)kb * Nc;
    const size_t bstep = (size_t)4 * Nc;

    v8f acc0 = {}, acc1 = {}, acc2 = {}, acc3 = {};
    for (int k0 = 0; k0 < K; k0 += 4) {
        v2f a;  a.x = ap[0];     a.y = ap[1];
        v2f b0; b0.x = bp[cc0];  b0.y = bp[Nc + cc0];
        v2f b1; b1.x = bp[cc1];  b1.y = bp[Nc + cc1];
        v2f b2; b2.x = bp[cc2];  b2.y = bp[Nc + cc2];
        v2f b3; b3.x = bp[cc3];  b3.y = bp[Nc + cc3];
        acc0 = __builtin_amdgcn_wmma_f32_16x16x4_f32(false, a, false, b0, (short)0, acc0, false, false);
        acc1 = __builtin_amdgcn_wmma_f32_16x16x4_f32(false, a, false, b1, (short)0, acc1, false, false);
        acc2 = __builtin_amdgcn_wmma_f32_16x16x4_f32(false, a, false, b2, (short)0, acc2, false, false);
        acc3 = __builtin_amdgcn_wmma_f32_16x16x4_f32(false, a, false, b3, (short)0, acc3, false, false);
        ap += 4;
        bp += bstep;
    }

    const int rbase = tm * 16 + ((lane >> 4) << 3);
    auto store_tile = [&](const v8f& acc, int j) {
        const int ncol = sn * 64 + 16 * j + l15;
        if (ncol >= Nc) return;
        const float bb = bias ? bias[ncol] : 0.f;
#pragma unroll
        for (int r = 0; r < 8; ++r) {
            const int mi = rbase + r;
            if (mi < M) {
                float v = acc[r] + bb;
                if (act == 1) v = v > 0.f ? v : 0.f;
                C[(size_t)mi * Nc + ncol] = v;
            }
        }
    };
    store_tile(acc0, 0);
    store_tile(acc1, 1);
    store_tile(acc2, 2);
    store_tile(acc3, 3);
}

// a_src[n,h] = sum_c h[n, h*64 + c] * att_src[h,c] ; same for a_dst
__global__ void att_scores(const float* __restrict__ h,
                           const float* __restrict__ att_src,
                           const float* __restrict__ att_dst,
                           float* __restrict__ asrc, float* __restrict__ adst,
                           int N)
{
    int idx = blockIdx.x * blockDim.x + threadIdx.x;
    if (idx >= N * GAT_H) return;
    int n = idx >> 2, hh = idx & 3;
    const float* hp = h + (size_t)n * GAT_HC + hh * GAT_C;
    const float* as = att_src + hh * GAT_C;
    const float* ad = att_dst + hh * GAT_C;
    float s = 0.f, d = 0.f;
#pragma unroll 4
    for (int c = 0; c < GAT_C; ++c) { float v = hp[c]; s += v * as[c]; d += v * ad[c]; }
    asrc[idx] = s; adst[idx] = d;
}

__global__ void init_softmax(unsigned* __restrict__ mkey, float* __restrict__ denom, int NH)
{
    int i = blockIdx.x * blockDim.x + threadIdx.x;
    if (i < NH) { mkey[i] = 0u; denom[i] = 0.f; }
}

__global__ void zero_f32(float* __restrict__ p, long long n)
{
    long long i = (long long)blockIdx.x * blockDim.x + threadIdx.x;
    if (i < n) p[i] = 0.f;
}

__device__ __forceinline__ float leaky(float e) { return e > 0.f ? e : NEG_SLOPE * e; }

// pass A: running max per (dst, head) over leaky(a_src[s]+a_dst[d])
__global__ void edge_max(const int* __restrict__ src, const int* __restrict__ dst,
                         int E, int N,
                         const float* __restrict__ asrc, const float* __restrict__ adst,
                         unsigned* __restrict__ mkey)
{
    long long t = (long long)blockIdx.x * blockDim.x + threadIdx.x;
    long long tot = (long long)(E + N) * GAT_H;
    if (t >= tot) return;
    int e = (int)(t >> 2), hh = (int)(t & 3);
    int s = (e < E) ? src[e] : (e - E);
    int d = (e < E) ? dst[e] : (e - E);
    float v = leaky(asrc[(s << 2) + hh] + adst[(d << 2) + hh]);
    atomicMax(&mkey[(d << 2) + hh], f2ord(v));
}

// pass B: ex = exp(e - m[d]), stash per-edge, accumulate denominator
__global__ void edge_expsum(const int* __restrict__ src, const int* __restrict__ dst,
                            int E, int N,
                            const float* __restrict__ asrc, const float* __restrict__ adst,
                            const unsigned* __restrict__ mkey,
                            float* __restrict__ exbuf, float* __restrict__ denom)
{
    long long t = (long long)blockIdx.x * blockDim.x + threadIdx.x;
    long long tot = (long long)(E + N) * GAT_H;
    if (t >= tot) return;
    int e = (int)(t >> 2), hh = (int)(t & 3);
    int s = (e < E) ? src[e] : (e - E);
    int d = (e < E) ? dst[e] : (e - E);
    float v = leaky(asrc[(s << 2) + hh] + adst[(d << 2) + hh]);
    float ex = expf(v - ord2f(mkey[(d << 2) + hh]));
    exbuf[t] = ex;
    atomicAdd(&denom[(d << 2) + hh], ex);
}

// pass C: agg[d,:] += h[s,:] * alpha(edge, head).  One wave per edge, 8 ch/lane.
__global__ void edge_aggregate(const int* __restrict__ src, const int* __restrict__ dst,
                               int E, int N,
                               const float* __restrict__ h,
                               const float* __restrict__ exbuf,
                               const float* __restrict__ denom,
                               float* __restrict__ agg)
{
    int lane = threadIdx.x & 31;
    long long e = (long long)blockIdx.x * (blockDim.x >> 5) + (threadIdx.x >> 5);
    if (e >= (long long)(E + N)) return;
    int s = (e < E) ? src[e] : (int)(e - E);
    int d = (e < E) ? dst[e] : (int)(e - E);
    const float* hs = h + (size_t)s * GAT_HC;
    float* ad = agg + (size_t)d * GAT_HC;
#pragma unroll
    for (int j = 0; j < GAT_HC / 32; ++j) {
        int c = lane + 32 * j;
        int hh = c >> 6;
        float alpha = exbuf[(e << 2) + hh] / (denom[(d << 2) + hh] + 1e-16f);
        atomicAdd(&ad[c], hs[c] * alpha);
    }
}

__global__ void bias_elu(float* __restrict__ x, const float* __restrict__ bias,
                         long long total)
{
    long long i = (long long)blockIdx.x * blockDim.x + threadIdx.x;
    if (i >= total) return;
    float v = x[i] + bias[(int)(i & (GAT_HC - 1))];
    x[i] = v > 0.f ? v : (expf(v) - 1.f);
}

// ---------------------------------------------------------------
extern "C" void kernel_launch(void* const* d_in, const int* in_sizes, int n_in,
                              void* d_out, int out_size, void* d_ws, size_t ws_size,
                              hipStream_t stream)
{
    const float* x    = (const float*)d_in[0];
    const int*   ei   = (const int*)d_in[1];
    const float* W1   = (const float*)d_in[2];
    const float* as1  = (const float*)d_in[3];
    const float* ad1  = (const float*)d_in[4];
    const float* b1   = (const float*)d_in[5];
    const float* W2   = (const float*)d_in[6];
    const float* as2  = (const float*)d_in[7];
    const float* ad2  = (const float*)d_in[8];
    const float* b2   = (const float*)d_in[9];
    const float* fcW1 = (const float*)d_in[10];
    const float* fcb1 = (const float*)d_in[11];
    const float* fcW2 = (const float*)d_in[12];
    const float* fcb2 = (const float*)d_in[13];
    float* out = (float*)d_out;

    const int N = 50000, E = 800000, FIN = 128, HID = 64, NCLS = 40;
    const int HC = GAT_HC; // 256
    const int* src = ei;
    const int* dst = ei + E;

    // workspace layout
    char* ws = (char*)d_ws;
    float*    hbuf  = (float*)ws;    ws += (size_t)N * HC * 4;        // GEMM output h
    float*    agg   = (float*)ws;    ws += (size_t)N * HC * 4;        // aggregation / layer input
    float*    asrc  = (float*)ws;    ws += (size_t)N * GAT_H * 4;
    float*    adst  = (float*)ws;    ws += (size_t)N * GAT_H * 4;
    unsigned* mkey  = (unsigned*)ws; ws += (size_t)N * GAT_H * 4;
    float*    denom = (float*)ws;    ws += (size_t)N * GAT_H * 4;
    float*    exbuf = (float*)ws;    ws += (size_t)(E + N) * GAT_H * 4;
    float*    fc1o  = exbuf;         // reused after softmax buffers are dead (N*HID <= (E+N)*H)

    auto gemm_blocks = [](int M, int Nc) {
        long long strips = (long long)((M + 15) / 16) * ((Nc + 63) / 64);
        return (int)((strips + 7) / 8);
    };
    const int NH = N * GAT_H;
    const long long ET = (long long)(E + N) * GAT_H;
    const int nhBlocks   = (NH + 255) / 256;
    const int edgeBlocks = (int)((ET + 255) / 256);
    const int aggBlocks  = (E + N + 7) / 8;        // 8 waves per 256-thread block
    const long long NHC  = (long long)N * HC;
    const int nhcBlocks  = (int)((NHC + 255) / 256);

    // -------------------- GAT layer 1 --------------------
    wmma_gemm_f32<<<gemm_blocks(N, HC), 256, 0, stream>>>(x, W1, nullptr, hbuf, N, FIN, HC, 0);
    att_scores<<<nhBlocks, 256, 0, stream>>>(hbuf, as1, ad1, asrc, adst, N);
    init_softmax<<<nhBlocks, 256, 0, stream>>>(mkey, denom, NH);
    zero_f32<<<nhcBlocks, 256, 0, stream>>>(agg, NHC);
    edge_max<<<edgeBlocks, 256, 0, stream>>>(src, dst, E, N, asrc, adst, mkey);
    edge_expsum<<<edgeBlocks, 256, 0, stream>>>(src, dst, E, N, asrc, adst, mkey, exbuf, denom);
    edge_aggregate<<<aggBlocks, 256, 0, stream>>>(src, dst, E, N, hbuf, exbuf, denom, agg);
    bias_elu<<<nhcBlocks, 256, 0, stream>>>(agg, b1, NHC);

    // -------------------- GAT layer 2 --------------------
    wmma_gemm_f32<<<gemm_blocks(N, HC), 256, 0, stream>>>(agg, W2, nullptr, hbuf, N, HC, HC, 0);
    att_scores<<<nhBlocks, 256, 0, stream>>>(hbuf, as2, ad2, asrc, adst, N);
    init_softmax<<<nhBlocks, 256, 0, stream>>>(mkey, denom, NH);
    zero_f32<<<nhcBlocks, 256, 0, stream>>>(agg, NHC);
    edge_max<<<edgeBlocks, 256, 0, stream>>>(src, dst, E, N, asrc, adst, mkey);
    edge_expsum<<<edgeBlocks, 256, 0, stream>>>(src, dst, E, N, asrc, adst, mkey, exbuf, denom);
    edge_aggregate<<<aggBlocks, 256, 0, stream>>>(src, dst, E, N, hbuf, exbuf, denom, agg);
    bias_elu<<<nhcBlocks, 256, 0, stream>>>(agg, b2, NHC);

    // -------------------- MLP head --------------------
    wmma_gemm_f32<<<gemm_blocks(N, HID), 256, 0, stream>>>(agg, fcW1, fcb1, fc1o, N, HC, HID, 1);
    wmma_gemm_f32<<<gemm_blocks(N, NCLS), 256, 0, stream>>>(fc1o, fcW2, fcb2, out, N, HID, NCLS, 0);
}